// AdaptiveSample_71605694759657
// MI455X (gfx1250) — compile-verified
//
#include <hip/hip_runtime.h>
#include <stdint.h>

// AdaptiveSample for MI455X (gfx1250), compile-only tuned.
// Memory-bound stencil/gather: stage feature/depth/guide tiles in LDS via the
// CDNA5 async global->LDS path (ASYNCcnt), serve the 15-tap gather from LDS.
// b128 DMA granules for all aligned traffic; b32 only for halo columns.

namespace {
constexpr int NB = 2, NC = 32, NH = 256, NW = 512;
constexpr int NS = 15, KSZ = 5, PADK = 2, KK = 25;
constexpr int T_H = 8, T_W = 32;              // 256 threads = 8x32 pixel tile
constexpr int HR  = T_H + 2 * PADK;           // 12 rows incl. halo
constexpr int WRW = 40;                       // padded row stride (16B-aligned interior)
constexpr int COFF = 4;                       // interior starts at column 4
// global col g maps to tile col cc = g - w0 + COFF; halo cc in {2,3,36,37}
constexpr int CSTR  = HR * WRW;               // 480 floats per channel
constexpr int FTILE = NC * CSTR;              // 15360 floats (61.4 KB)
constexpr int DTILE = HR * WRW;               // 480 floats
constexpr int GROW  = T_W * KK;               // 800 floats per guide row
constexpr int GTILE = T_H * GROW;             // 6400 floats (25.6 KB)
constexpr int NTHR  = 256;
constexpr float DEPTH_MAX = 192.0f;
}

// Low 32 bits of a flat LDS pointer == LDS-base-relative byte offset
// (flat aperture uses the low 32 address bits); matches async-load VDST
// semantics (VDST holds an LDS-base-relative byte address).
__device__ __forceinline__ unsigned lds_off(const void* p) {
  return (unsigned)(uintptr_t)p;
}

// GLOBAL_LOAD_ASYNC_TO_LDS_*, GVS mode: saddr(64b SGPR pair) + vaddr(32b offset).
__device__ __forceinline__ void async_ld_b32(unsigned lds_byte, const void* sbase,
                                             unsigned gbyte) {
  asm volatile("global_load_async_to_lds_b32 %0, %1, %2"
               :: "v"(lds_byte), "v"(gbyte), "s"(sbase)
               : "memory");
}
__device__ __forceinline__ void async_ld_b128(unsigned lds_byte, const void* sbase,
                                              unsigned gbyte) {
  asm volatile("global_load_async_to_lds_b128 %0, %1, %2"
               :: "v"(lds_byte), "v"(gbyte), "s"(sbase)
               : "memory");
}
__device__ __forceinline__ void wait_async0() {
  asm volatile("s_wait_asynccnt 0x0" ::: "memory");
}
__device__ __forceinline__ void wait_ds0() {
  asm volatile("s_wait_dscnt 0x0" ::: "memory");
}

__global__ __launch_bounds__(NTHR) void AdaptiveSample_kernel(
    const float* __restrict__ depth,   // [B,1,H,W]
    const float* __restrict__ feat,    // [B,C,H,W]
    const float* __restrict__ guide,   // [B,H,W,25]
    const int*   __restrict__ sidx,    // [15]
    float*       __restrict__ out)     // [B,C,H,W] out  ++  [B,C,H,W] feat copy
{
  __shared__ __align__(16) float ftile[FTILE];
  __shared__ __align__(16) float dtile[DTILE];
  __shared__ __align__(16) float gtile[GTILE];

  const int tid = (int)threadIdx.x;
  const int w0  = (int)blockIdx.x * T_W;
  const int h0  = (int)blockIdx.y * T_H;
  const int b   = (int)blockIdx.z;

  // ---- zero tiles that have halo / unused columns (b128 ds stores) ----
  {
    const float4 z4 = make_float4(0.f, 0.f, 0.f, 0.f);
    float4* f4 = (float4*)ftile;
    float4* d4 = (float4*)dtile;
#pragma unroll
    for (int p = 0; p < FTILE / 4 / NTHR; ++p) f4[tid + p * NTHR] = z4;  // 15 stores
    if (tid < DTILE / 4) d4[tid] = z4;
  }
  wait_ds0();
  __syncthreads();   // zeros visible before async DMA overwrites interior

  const unsigned f_l = lds_off(ftile);
  const unsigned d_l = lds_off(dtile);
  const unsigned g_l = lds_off(gtile);

  const float* fbase = feat  + (size_t)b * NC * NH * NW;
  const float* dbase = depth + (size_t)b * NH * NW;
  const float* gbase = guide + (((size_t)b * NH + h0) * NW + w0) * KK;

  // ---- features interior: 384 (c,r) rows x 8 b128 lanes (16B aligned) ----
  {
    const int seg = tid & 7;          // 4-float segment within the 32-wide row
    const int g0  = tid >> 3;         // (c,r) group, 32 per pass
#pragma unroll
    for (int p = 0; p < 12; ++p) {    // 12 * 32 = 384 rows
      int g  = g0 + p * 32;
      int c  = g / HR;
      int r  = g % HR;
      int gr = h0 + r - PADK;
      if ((unsigned)gr < (unsigned)NH)
        async_ld_b128(f_l + 4u * (unsigned)(c * CSTR + r * WRW + COFF) + 16u * seg,
                      fbase,
                      (unsigned)(4 * ((c * NH + gr) * NW + w0)) + 16u * seg);
    }
  }
  // ---- features halo: 384 rows x 4 masked b32 columns ----
  {
    const int j  = tid & 3;                         // 0,1 -> left; 2,3 -> right
    const int g0 = tid >> 2;                        // 64 rows per pass
    const int gc = w0 + ((j < 2) ? (j - 2) : (j + 30));   // w0-2,w0-1,w0+32,w0+33
    const int cc = (j < 2) ? (j + 2) : (j + 34);          // 2,3,36,37
#pragma unroll
    for (int p = 0; p < 6; ++p) {                   // 6 * 64 = 384 rows
      int g  = g0 + p * 64;
      int c  = g / HR;
      int r  = g % HR;
      int gr = h0 + r - PADK;
      if ((unsigned)gr < (unsigned)NH && (unsigned)gc < (unsigned)NW)
        async_ld_b32(f_l + 4u * (unsigned)(c * CSTR + r * WRW + cc),
                     fbase, (unsigned)(4 * ((c * NH + gr) * NW + gc)));
    }
  }
  // ---- depth tile: 12 rows, interior b128 + masked b32 halo ----
  if (tid < HR * 8) {                 // interior: 12 rows x 8 lanes
    int r = tid >> 3, seg = tid & 7;
    int gr = h0 + r - PADK;
    if ((unsigned)gr < (unsigned)NH)
      async_ld_b128(d_l + 4u * (unsigned)(r * WRW + COFF) + 16u * seg,
                    dbase, (unsigned)(4 * (gr * NW + w0)) + 16u * seg);
  } else if (tid < HR * 8 + HR * 4) { // halo: 12 rows x 4 cols
    int t = tid - HR * 8;
    int r = t >> 2, j = t & 3;
    int gr = h0 + r - PADK;
    int gc = w0 + ((j < 2) ? (j - 2) : (j + 30));
    int cc = (j < 2) ? (j + 2) : (j + 34);
    if ((unsigned)gr < (unsigned)NH && (unsigned)gc < (unsigned)NW)
      async_ld_b32(d_l + 4u * (unsigned)(r * WRW + cc),
                   dbase, (unsigned)(4 * (gr * NW + gc)));
  }
  // ---- guide tile: fully contiguous rows, pure b128 (no zeroing needed) ----
  {
#pragma unroll
    for (int p = 0; p < 7; ++p) {     // 1600 b128 slots
      int i = tid + p * NTHR;
      if (i < GTILE / 4) {
        int rr = i / (GROW / 4);
        int jj = i % (GROW / 4);
        async_ld_b128(g_l + 16u * (unsigned)i, gbase,
                      (unsigned)(4 * (rr * NW * KK) + 16 * jj));
      }
    }
  }

  wait_async0();
  __syncthreads();

  // ---- per-sample metadata: kernel index, LDS shift, Gaussian prior ----
  int   kidx[NS], soff[NS];
  float posw[NS];
  float psum = 0.0f;
#pragma unroll
  for (int s = 0; s < NS; ++s) {
    int k = sidx[s];
    kidx[s] = k;
    int dy = k / KSZ, dx = k % KSZ;
    soff[s] = dy * WRW + dx;
    float fx = (float)dx - 2.0f, fy = (float)dy - 2.0f;
    float pw = __expf(-0.5f * __fsqrt_rn(fx * fx + fy * fy));
    posw[s] = pw;
    psum += pw;
  }
  const float pnorm = __frcp_rn(psum);

  // ---- per-pixel weights + softmax (invalid taps keep raw=0 -> exp mass) ----
  const int ty = tid >> 5, tx = tid & 31;   // wave32 == one row of 32 pixels
  const int hh = h0 + ty,  ww = w0 + tx;
  const int pbase = ty * WRW + tx + (COFF - PADK);  // read idx = pbase + soff[s]
  const int gb    = (ty * T_W + tx) * KK;

  float wgt[NS];
  float m = -3.4e38f;
#pragma unroll
  for (int s = 0; s < NS; ++s) {
    float d = dtile[pbase + soff[s]];
    float v = (d > 0.0f && d < DEPTH_MAX) ? 1.0f : 0.0f;
    float r = v * (posw[s] * pnorm) * gtile[gb + kidx[s]];
    wgt[s] = r;
    m = fmaxf(m, r);
  }
  float esum = 0.0f;
#pragma unroll
  for (int s = 0; s < NS; ++s) {
    float e = __expf(wgt[s] - m);
    wgt[s] = e;
    esum += e;
  }
  const float einv = __frcp_rn(esum);
#pragma unroll
  for (int s = 0; s < NS; ++s) wgt[s] *= einv;

  // ---- weighted 15-tap gather per channel + features passthrough ----
  float* outp = out + ((size_t)b * NC * NH + (size_t)hh) * NW + ww;
  float* fcpy = outp + (size_t)NB * NC * NH * NW;
  const int ctr = PADK * WRW + PADK;   // center tap offset (dy=dx=2)
#pragma unroll 4
  for (int c = 0; c < NC; ++c) {
    const int fb = c * CSTR + pbase;
    float acc = 0.0f;
#pragma unroll
    for (int s = 0; s < NS; ++s) acc += ftile[fb + soff[s]] * wgt[s];
    size_t o = (size_t)c * (NH * NW);
    outp[o] = acc;                 // coalesced 128B per wave
    fcpy[o] = ftile[fb + ctr];     // passthrough from LDS interior
  }
}

extern "C" void kernel_launch(void* const* d_in, const int* in_sizes, int n_in,
                              void* d_out, int out_size, void* d_ws, size_t ws_size,
                              hipStream_t stream) {
  const float* depth = (const float*)d_in[0];
  const float* feat  = (const float*)d_in[1];
  const float* guide = (const float*)d_in[2];
  const int*   sidx  = (const int*)d_in[3];
  float* out = (float*)d_out;
  (void)in_sizes; (void)n_in; (void)out_size; (void)d_ws; (void)ws_size;

  dim3 grid(NW / T_W, NH / T_H, NB);   // 16 x 32 x 2 = 1024 blocks
  hipLaunchKernelGGL(AdaptiveSample_kernel, grid, dim3(NTHR), 0, stream,
                     depth, feat, guide, sidx, out);
}